// ImprovedVoiceCloningModel_42743514530556
// MI455X (gfx1250) — compile-verified
//
#include <hip/hip_runtime.h>
#include <math.h>

// ---------------------------------------------------------------------------
// Model constants (from reference)
// ---------------------------------------------------------------------------
#define TSTEPS 1000
#define NMELS  80
#define HID    1024
#define G4     4096      // 4*HID
#define PRE    256
#define PF     512

typedef __attribute__((ext_vector_type(2))) float v2f;
typedef __attribute__((ext_vector_type(8))) float v8f;

// ---------------------------------------------------------------------------
// fp32 WMMA GEMM:  OUT[M,N] = act(A[M,K] @ W[N,K]^T + bias[N])
// One wave per 16x16 tile, K-loop in steps of 4 using V_WMMA_F32_16X16X4_F32.
// M-edge: A load rows are CLAMPED to M-1 (in-bounds, divergence-free); the
// garbage they produce only lands in D rows >= M, which are never stored.
//   A layout (16x4 f32): lanes 0-15 rows, vgpr0/1 = K(2h)/K(2h+1)
//   B layout (4x16 f32): lanes 0-15 = K 0..1, lanes 16-31 = K 2..3
//   C/D: vgpr v, lanes 0-15 -> M=v, lanes 16-31 -> M=v+8
// ---------------------------------------------------------------------------
__global__ __launch_bounds__(32)
void gemm_wmma_kernel(const float* __restrict__ A, int lda,
                      const float* __restrict__ W, int ldw,
                      const float* __restrict__ bias,
                      float* __restrict__ OUT, int ldo,
                      int M, int N, int K, int act /*0 none, 1 relu*/)
{
    const int lane = threadIdx.x;
    const int r    = lane & 15;
    const int h    = lane >> 4;
    const int row  = blockIdx.y * 16 + r;            // A row this lane feeds
    const int rowl = (row < M) ? row : (M - 1);      // clamp: stay in-bounds
    const int col  = blockIdx.x * 16 + r;            // W row (output col)
    const float* __restrict__ Arow = A + (size_t)rowl * lda + 2 * h;
    const float* __restrict__ Wrow = W + (size_t)col * ldw + 2 * h;

    v8f c = {};
#pragma unroll 2
    for (int kk = 0; kk < K; kk += 4) {
        const v2f a = *(const v2f*)(Arow + kk);
        const v2f b = *(const v2f*)(Wrow + kk);
        c = __builtin_amdgcn_wmma_f32_16x16x4_f32(false, a, false, b,
                                                  (short)0, c, false, false);
    }

    const float bval = bias ? bias[col] : 0.f;
#pragma unroll
    for (int v = 0; v < 8; ++v) {
        const int rowo = blockIdx.y * 16 + v + 8 * h;
        if (rowo < M) {
            float val = c[v] + bval;
            if (act == 1) val = fmaxf(val, 0.f);
            OUT[(size_t)rowo * ldo + col] = val;
        }
    }
}

// ---------------------------------------------------------------------------
// Postnet conv1d(k=5, pad=2) + BatchNorm + optional tanh as a WMMA GEMM with
// K = 5*Cin (implicit im2col).  IN points at a PADDED buffer whose row 0 is
// logical time -2 and whose 2-row halos are pre-zeroed, so boundary taps need
// no predication; tap loop hoisted (no divisions).  OUT points at logical
// row 0 of the next layer's padded buffer.
// ---------------------------------------------------------------------------
__global__ __launch_bounds__(32)
void postconv_wmma_kernel(const float* __restrict__ INp, int Cin,
                          const float* __restrict__ Wt,       // (Cout, Cin, 5)
                          const float* __restrict__ bconv,
                          const float* __restrict__ bn_g,
                          const float* __restrict__ bn_be,
                          const float* __restrict__ bn_m,
                          const float* __restrict__ bn_v,
                          float* __restrict__ OUT, int Cout,
                          int T, int do_tanh)
{
    const int lane = threadIdx.x;
    const int r    = lane & 15;
    const int h    = lane >> 4;
    const int row  = blockIdx.y * 16 + r;            // time index
    const int rowl = (row < T) ? row : (T - 1);      // clamp tile overhang
    const int col  = blockIdx.x * 16 + r;            // output channel
    const float* __restrict__ Wrow = Wt + (size_t)col * Cin * 5;

    v8f c = {};
#pragma unroll
    for (int d = 0; d < 5; ++d) {
        // padded index: logical (rowl + d - 2) + 2 halo rows = rowl + d
        const float* __restrict__ ipA = INp + (size_t)(rowl + d) * Cin + 2 * h;
        const float* __restrict__ wpB = Wrow + (size_t)(2 * h) * 5 + d;
        for (int kk = 0; kk < Cin; kk += 4) {
            const v2f a = *(const v2f*)(ipA + kk);
            v2f b;
            b.x = wpB[(size_t)kk * 5];
            b.y = wpB[(size_t)kk * 5 + 5];
            c = __builtin_amdgcn_wmma_f32_16x16x4_f32(false, a, false, b,
                                                      (short)0, c, false, false);
        }
    }

    const float bb    = bconv[col];
    const float scale = bn_g[col] * rsqrtf(bn_v[col] + 1e-5f);
    const float shift = bn_be[col] - bn_m[col] * scale;
#pragma unroll
    for (int v = 0; v < 8; ++v) {
        const int rowo = blockIdx.y * 16 + v + 8 * h;
        if (rowo < T) {
            float val = (c[v] + bb) * scale + shift;
            if (do_tanh) val = tanhf(val);
            OUT[(size_t)rowo * Cout + col] = val;
        }
    }
}

// ---------------------------------------------------------------------------
// Zero the 2-row halos of the padded conv activation buffers.
// ---------------------------------------------------------------------------
__global__ void zero_pads_kernel(float* __restrict__ melp,
                                 float* __restrict__ pap,
                                 float* __restrict__ pbp)
{
    const int i = blockIdx.x * blockDim.x + threadIdx.x;
    if (i < 2 * NMELS) {
        melp[i] = 0.f;
        melp[(size_t)(TSTEPS + 2) * NMELS + i] = 0.f;
    }
    if (i < 2 * PF) {
        pap[i] = 0.f;  pap[(size_t)(TSTEPS + 2) * PF + i] = 0.f;
        pbp[i] = 0.f;  pbp[(size_t)(TSTEPS + 2) * PF + i] = 0.f;
    }
}

// ---------------------------------------------------------------------------
// One-time prep: transpose the three recurrent 4096x1024 matrices to
// K-major (coalesced streaming in the scan), fold the (constant!) attention
// context = combined_embedding into LSTM1's input bias and the projection
// bias, reset the grid-barrier counters.
// ---------------------------------------------------------------------------
__global__ void prep_kernel(const float* __restrict__ whh1,
                            const float* __restrict__ wih2,
                            const float* __restrict__ whh2,
                            const float* __restrict__ wih1,   // (4096, 1280)
                            const float* __restrict__ emb,    // (1024)
                            const float* __restrict__ bih1, const float* __restrict__ bhh1,
                            const float* __restrict__ bih2, const float* __restrict__ bhh2,
                            const float* __restrict__ proj_w, // (80, 2048)
                            const float* __restrict__ proj_b,
                            float* __restrict__ whh1T, float* __restrict__ wih2T,
                            float* __restrict__ whh2T,
                            float* __restrict__ c1bias, float* __restrict__ c1emb,
                            float* __restrict__ c2bias, float* __restrict__ projc,
                            unsigned* __restrict__ bar)
{
    const size_t gid    = (size_t)blockIdx.x * blockDim.x + threadIdx.x;
    const size_t stride = (size_t)gridDim.x * blockDim.x;

    for (size_t idx = gid; idx < (size_t)G4 * HID; idx += stride) {
        const size_t k = idx >> 12;         // 0..1023
        const size_t j = idx & 4095;        // 0..4095
        const size_t src = j * HID + k;
        whh1T[idx] = whh1[src];
        wih2T[idx] = wih2[src];
        whh2T[idx] = whh2[src];
    }
    for (size_t j = gid; j < G4; j += stride) {
        float s = 0.f;
        const float* wrow = wih1 + j * (PRE + HID) + PRE;  // ctx columns
        for (int e = 0; e < HID; ++e) s += emb[e] * wrow[e];
        c1emb[j]  = s;
        c1bias[j] = bih1[j] + bhh1[j];
        c2bias[j] = bih2[j] + bhh2[j];
    }
    for (size_t n = gid; n < NMELS; n += stride) {
        float s = proj_b[n];
        const float* wrow = proj_w + n * (HID + HID) + HID; // ctx columns
        for (int e = 0; e < HID; ++e) s += emb[e] * wrow[e];
        projc[n] = s;
    }
    if (gid == 0) { bar[0] = 0u; bar[1] = 0u; }
}

// ---------------------------------------------------------------------------
// Persistent sequential LSTM scan.  64 WGs x 256 threads; split-K x4.
// Weight reads are coalesced (K-major layout, 48 MB total -> fully
// L2-resident; ~48 GB streamed over 1000 steps = the roofline).  Two grid
// barriers per step; each WG redundantly (bit-exactly) recomputes the cheap
// cell updates from the shared split-K partials, so h/c state lives in its
// own LDS with no extra synchronization.
// ---------------------------------------------------------------------------
#define SCAN_NWG 64

__device__ __forceinline__ float sigf(float x) { return 1.f / (1.f + expf(-x)); }

__device__ __forceinline__ void grid_barrier(unsigned* cnt, unsigned* gen)
{
    __threadfence();                 // publish this block's stores (device scope)
    __syncthreads();
    if (threadIdx.x == 0) {
        const unsigned g = __hip_atomic_load(gen, __ATOMIC_RELAXED,
                                             __HIP_MEMORY_SCOPE_AGENT);
        const unsigned prev = __hip_atomic_fetch_add(cnt, 1u, __ATOMIC_ACQ_REL,
                                                     __HIP_MEMORY_SCOPE_AGENT);
        if (prev == SCAN_NWG - 1u) {
            __hip_atomic_store(cnt, 0u, __ATOMIC_RELAXED, __HIP_MEMORY_SCOPE_AGENT);
            __hip_atomic_fetch_add(gen, 1u, __ATOMIC_ACQ_REL,
                                   __HIP_MEMORY_SCOPE_AGENT);
        } else {
            while (__hip_atomic_load(gen, __ATOMIC_ACQUIRE,
                                     __HIP_MEMORY_SCOPE_AGENT) == g) {
                __builtin_amdgcn_s_sleep(2);
            }
        }
    }
    __syncthreads();
    __threadfence();                 // acquire: see other blocks' stores
}

__global__ __launch_bounds__(256)
void scan_kernel(const float* __restrict__ X1,       // (1000, 4096)
                 const float* __restrict__ c1bias,
                 const float* __restrict__ c1emb,
                 const float* __restrict__ c2bias,
                 const float* __restrict__ whh1T,    // (1024, 4096) K-major
                 const float* __restrict__ wih2T,
                 const float* __restrict__ whh2T,
                 const float* __restrict__ init_h,   // (2, 1024)
                 const float* __restrict__ init_c,
                 float* __restrict__ z1part,         // (4, 4096)
                 float* __restrict__ z2part,
                 float* __restrict__ H2,             // (1000, 1024)
                 unsigned* __restrict__ bar)
{
    __shared__ float h1[HID], c1[HID], h2[HID], c2[HID];
    const int tid = threadIdx.x;
    for (int m = tid; m < HID; m += 256) {
        h1[m] = init_h[m]; h2[m] = init_h[HID + m];
        c1[m] = init_c[m]; c2[m] = init_c[HID + m];
    }
    __syncthreads();

    const int gtid  = blockIdx.x * 256 + tid;
    const int j     = gtid & 4095;
    const int chunk = gtid >> 12;           // 0..3
    const int k0    = chunk << 8;           // 256-wide K chunk
    const float* __restrict__ W1  = whh1T + (size_t)k0 * G4 + j;
    const float* __restrict__ W2a = wih2T + (size_t)k0 * G4 + j;
    const float* __restrict__ W2b = whh2T + (size_t)k0 * G4 + j;

    for (int t = 0; t < TSTEPS; ++t) {
        // ---- phase 1: z1 split-K partials: h1 @ whh1^T -------------------
        {
            float acc = 0.f;
            const float* wp = W1;
#pragma unroll 4
            for (int k = 0; k < 256; ++k) { acc += h1[k0 + k] * wp[0]; wp += G4; }
            z1part[(chunk << 12) + j] = acc;
        }
        grid_barrier(bar, bar + 1);

        // ---- phase 2: combine + LSTM cell 1 (redundant per WG, bit-exact) -
        {
            const float* x1t = X1 + (size_t)t * G4;
            const float cemb = (t > 0) ? 1.f : 0.f;   // ctx==0 only at t==0
            for (int m = tid; m < HID; m += 256) {
                int i0 = m, i1 = m + HID, i2 = m + 2 * HID, i3 = m + 3 * HID;
                float zi = c1bias[i0] + cemb * c1emb[i0] + x1t[i0]
                         + z1part[i0] + z1part[4096 + i0] + z1part[8192 + i0] + z1part[12288 + i0];
                float zf = c1bias[i1] + cemb * c1emb[i1] + x1t[i1]
                         + z1part[i1] + z1part[4096 + i1] + z1part[8192 + i1] + z1part[12288 + i1];
                float zg = c1bias[i2] + cemb * c1emb[i2] + x1t[i2]
                         + z1part[i2] + z1part[4096 + i2] + z1part[8192 + i2] + z1part[12288 + i2];
                float zo = c1bias[i3] + cemb * c1emb[i3] + x1t[i3]
                         + z1part[i3] + z1part[4096 + i3] + z1part[8192 + i3] + z1part[12288 + i3];
                float cn = sigf(zf) * c1[m] + sigf(zi) * tanhf(zg);
                c1[m] = cn;
                h1[m] = sigf(zo) * tanhf(cn);
            }
        }
        __syncthreads();

        // ---- phase 2b: z2 partials: h1 @ wih2^T + h2 @ whh2^T ------------
        {
            float acc = 0.f;
            const float* wa = W2a;
            const float* wb = W2b;
#pragma unroll 4
            for (int k = 0; k < 256; ++k) {
                acc += h1[k0 + k] * wa[0] + h2[k0 + k] * wb[0];
                wa += G4; wb += G4;
            }
            z2part[(chunk << 12) + j] = acc;
        }
        grid_barrier(bar, bar + 1);

        // ---- phase 3: combine + LSTM cell 2; WG0 records h2 history ------
        {
            for (int m = tid; m < HID; m += 256) {
                int i0 = m, i1 = m + HID, i2 = m + 2 * HID, i3 = m + 3 * HID;
                float zi = c2bias[i0] + z2part[i0] + z2part[4096 + i0] + z2part[8192 + i0] + z2part[12288 + i0];
                float zf = c2bias[i1] + z2part[i1] + z2part[4096 + i1] + z2part[8192 + i1] + z2part[12288 + i1];
                float zg = c2bias[i2] + z2part[i2] + z2part[4096 + i2] + z2part[8192 + i2] + z2part[12288 + i2];
                float zo = c2bias[i3] + z2part[i3] + z2part[4096 + i3] + z2part[8192 + i3] + z2part[12288 + i3];
                float cn = sigf(zf) * c2[m] + sigf(zi) * tanhf(zg);
                c2[m] = cn;
                float hn = sigf(zo) * tanhf(cn);
                h2[m] = hn;
                if (blockIdx.x == 0) H2[(size_t)t * HID + m] = hn;
            }
        }
        __syncthreads();
    }
}

// ---------------------------------------------------------------------------
// out[0:80000]     = mel + postnet(mel)
// out[80000:81000] = alignments = softmax over a length-1 axis == 1.0
// ---------------------------------------------------------------------------
__global__ void finalize_kernel(const float* __restrict__ mel,
                                const float* __restrict__ post,
                                float* __restrict__ out)
{
    const int i = blockIdx.x * blockDim.x + threadIdx.x;
    if (i < TSTEPS * NMELS)               out[i] = mel[i] + post[i];
    else if (i < TSTEPS * NMELS + TSTEPS) out[i] = 1.0f;
}

// ---------------------------------------------------------------------------
// Host launch
// ---------------------------------------------------------------------------
extern "C" void kernel_launch(void* const* d_in, const int* in_sizes, int n_in,
                              void* d_out, int out_size, void* d_ws, size_t ws_size,
                              hipStream_t stream)
{
    (void)n_in; (void)out_size; (void)ws_size;
    // Detect flattening order: insertion order => in_sizes[1] == target_mels
    const bool ins = (in_sizes[1] == TSTEPS * NMELS);

    int I_emb, I_mels, I_pw1, I_pb1, I_pw2, I_pb2, I_wih1, I_whh1, I_bih1, I_bhh1,
        I_wih2, I_whh2, I_bih2, I_bhh2, I_projw, I_projb, I_inith, I_initc;
    int I_pw[5], I_pbias[5], I_pg[5], I_pbe[5], I_pm[5], I_pv[5];
    if (ins) {
        I_emb = 0; I_mels = 1; I_pw1 = 2; I_pb1 = 3; I_pw2 = 4; I_pb2 = 5;
        I_wih1 = 6; I_whh1 = 7; I_bih1 = 8; I_bhh1 = 9;
        I_wih2 = 10; I_whh2 = 11; I_bih2 = 12; I_bhh2 = 13;
        I_projw = 19; I_projb = 20;
        for (int l = 0; l < 5; ++l) {
            const int b = 21 + 6 * l;
            I_pw[l] = b; I_pbias[l] = b + 1; I_pg[l] = b + 2;
            I_pbe[l] = b + 3; I_pm[l] = b + 4; I_pv[l] = b + 5;
        }
        I_inith = 51; I_initc = 52;
    } else {  // alphabetical pytree order
        I_emb = 0; I_bhh1 = 1; I_bhh2 = 2; I_bih1 = 3; I_bih2 = 4;
        I_initc = 5; I_inith = 6;
        for (int l = 0; l < 5; ++l) {
            const int b = 10 + 6 * l;
            I_pbias[l] = b; I_pbe[l] = b + 1; I_pg[l] = b + 2;
            I_pm[l] = b + 3; I_pv[l] = b + 4; I_pw[l] = b + 5;
        }
        I_pb1 = 40; I_pb2 = 41; I_pw1 = 42; I_pw2 = 43;
        I_projb = 44; I_projw = 45;
        I_whh1 = 48; I_whh2 = 49; I_wih1 = 50; I_wih2 = 51; I_mels = 52;
    }

    const float* emb   = (const float*)d_in[I_emb];
    const float* mels  = (const float*)d_in[I_mels];
    const float* pw1   = (const float*)d_in[I_pw1];
    const float* pb1   = (const float*)d_in[I_pb1];
    const float* pw2   = (const float*)d_in[I_pw2];
    const float* pb2   = (const float*)d_in[I_pb2];
    const float* wih1  = (const float*)d_in[I_wih1];
    const float* whh1  = (const float*)d_in[I_whh1];
    const float* bih1  = (const float*)d_in[I_bih1];
    const float* bhh1  = (const float*)d_in[I_bhh1];
    const float* wih2  = (const float*)d_in[I_wih2];
    const float* whh2  = (const float*)d_in[I_whh2];
    const float* bih2  = (const float*)d_in[I_bih2];
    const float* bhh2  = (const float*)d_in[I_bhh2];
    const float* projw = (const float*)d_in[I_projw];
    const float* projb = (const float*)d_in[I_projb];
    const float* inith = (const float*)d_in[I_inith];
    const float* initc = (const float*)d_in[I_initc];

    // Workspace carve-up (floats); ~78 MB total
    float* w = (float*)d_ws;
    size_t off = 0;
    auto alloc = [&](size_t n) { float* p = w + off; off += n; return p; };
    float* whh1T = alloc((size_t)HID * G4);
    float* wih2T = alloc((size_t)HID * G4);
    float* whh2T = alloc((size_t)HID * G4);
    float* X1    = alloc((size_t)TSTEPS * G4);
    float* H2    = alloc((size_t)TSTEPS * HID);
    float* P1    = alloc((size_t)TSTEPS * PRE);   // prenet l1 out; reused for POST
    float* P2    = alloc((size_t)TSTEPS * PRE);
    float* MELp  = alloc((size_t)(TSTEPS + 4) * NMELS);  // padded: row0 = t=-2
    float* PAp   = alloc((size_t)(TSTEPS + 4) * PF);
    float* PBp   = alloc((size_t)(TSTEPS + 4) * PF);
    float* z1p   = alloc(4 * G4);
    float* z2p   = alloc(4 * G4);
    float* c1b   = alloc(G4);
    float* c1e   = alloc(G4);
    float* c2b   = alloc(G4);
    float* projc = alloc(128);
    unsigned* bar = (unsigned*)alloc(64);

    float* MEL  = MELp + 2 * NMELS;   // logical t=0
    float* POST = P1;                 // unpadded (final layer output)

    const int mtiles = (TSTEPS + 15) / 16;   // 63

    // 1) prep: transposes + folded-context biases + barrier reset + halos
    prep_kernel<<<2048, 256, 0, stream>>>(whh1, wih2, whh2, wih1, emb,
                                          bih1, bhh1, bih2, bhh2,
                                          projw, projb,
                                          whh1T, wih2T, whh2T,
                                          c1b, c1e, c2b, projc, bar);
    zero_pads_kernel<<<4, 256, 0, stream>>>(MELp, PAp, PBp);

    // 2) prenet (two relu GEMMs), WMMA fp32
    gemm_wmma_kernel<<<dim3(PRE / 16, mtiles), 32, 0, stream>>>(
        mels, NMELS, pw1, NMELS, pb1, P1, PRE, TSTEPS, PRE, NMELS, 1);
    gemm_wmma_kernel<<<dim3(PRE / 16, mtiles), 32, 0, stream>>>(
        P1, PRE, pw2, PRE, pb2, P2, PRE, TSTEPS, PRE, PRE, 1);

    // 3) X1 = prenet_out @ wih1[:, :256]^T  (teacher-forced -> fully parallel)
    gemm_wmma_kernel<<<dim3(G4 / 16, mtiles), 32, 0, stream>>>(
        P2, PRE, wih1, PRE + HID, nullptr, X1, G4, TSTEPS, G4, PRE, 0);

    // 4) persistent sequential LSTM scan (roofline: 48 GB L2-resident weights)
    scan_kernel<<<SCAN_NWG, 256, 0, stream>>>(X1, c1b, c1e, c2b,
                                              whh1T, wih2T, whh2T,
                                              inith, initc, z1p, z2p, H2, bar);

    // 5) mel = H2 @ proj_w[:, :1024]^T + (proj_b + ctx part)  -> padded MEL
    gemm_wmma_kernel<<<dim3(NMELS / 16, mtiles), 32, 0, stream>>>(
        H2, HID, projw, 2 * HID, projc, MEL, NMELS, TSTEPS, NMELS, HID, 0);

    // 6) postnet: conv5+BN+tanh x4, conv5+BN   (padded in, padded out)
    postconv_wmma_kernel<<<dim3(PF / 16, mtiles), 32, 0, stream>>>(
        MELp, NMELS, (const float*)d_in[I_pw[0]], (const float*)d_in[I_pbias[0]],
        (const float*)d_in[I_pg[0]], (const float*)d_in[I_pbe[0]],
        (const float*)d_in[I_pm[0]], (const float*)d_in[I_pv[0]],
        PAp + 2 * PF, PF, TSTEPS, 1);
    postconv_wmma_kernel<<<dim3(PF / 16, mtiles), 32, 0, stream>>>(
        PAp, PF, (const float*)d_in[I_pw[1]], (const float*)d_in[I_pbias[1]],
        (const float*)d_in[I_pg[1]], (const float*)d_in[I_pbe[1]],
        (const float*)d_in[I_pm[1]], (const float*)d_in[I_pv[1]],
        PBp + 2 * PF, PF, TSTEPS, 1);
    postconv_wmma_kernel<<<dim3(PF / 16, mtiles), 32, 0, stream>>>(
        PBp, PF, (const float*)d_in[I_pw[2]], (const float*)d_in[I_pbias[2]],
        (const float*)d_in[I_pg[2]], (const float*)d_in[I_pbe[2]],
        (const float*)d_in[I_pm[2]], (const float*)d_in[I_pv[2]],
        PAp + 2 * PF, PF, TSTEPS, 1);
    postconv_wmma_kernel<<<dim3(PF / 16, mtiles), 32, 0, stream>>>(
        PAp, PF, (const float*)d_in[I_pw[3]], (const float*)d_in[I_pbias[3]],
        (const float*)d_in[I_pg[3]], (const float*)d_in[I_pbe[3]],
        (const float*)d_in[I_pm[3]], (const float*)d_in[I_pv[3]],
        PBp + 2 * PF, PF, TSTEPS, 1);
    postconv_wmma_kernel<<<dim3(NMELS / 16, mtiles), 32, 0, stream>>>(
        PBp, PF, (const float*)d_in[I_pw[4]], (const float*)d_in[I_pbias[4]],
        (const float*)d_in[I_pg[4]], (const float*)d_in[I_pbe[4]],
        (const float*)d_in[I_pm[4]], (const float*)d_in[I_pv[4]],
        POST, NMELS, TSTEPS, 0);

    // 7) out = mel + post ; alignments = 1.0
    const int ntot = TSTEPS * NMELS + TSTEPS;
    finalize_kernel<<<(ntot + 255) / 256, 256, 0, stream>>>(MEL, POST, (float*)d_out);
}